// TN_62371515072790
// MI455X (gfx1250) — compile-verified
//
#include <hip/hip_runtime.h>
#include <stdint.h>

// MPS chain contraction for MI455X (gfx1250):
//   v <- v @ core[x[s]]  for s in [0,1024), then out = dot(v, right).
// Roofline: 2.1 GFLOP vs 4 GB of matrix reads (0.5 FLOP/B); the 32 cores
// (128 MB) fit in the 192 MB L2 -> strictly L2-bandwidth + sync-latency bound.
// Design: one persistent dispatch (64 WGs x 256 thr), device-scope barrier per
// step, TDM async panel prefetch into double-buffered LDS (TENSORcnt), and the
// per-WG 1x1024 @ 1024x16 GEMV tile on V_WMMA_F32_16X16X4_F32 with 4
// independent accumulators so LDS load latency overlaps the matrix pipe.

#define D_DIM   1024
#define F_LEN   1024
#define NWG     64
#define COLS    16                 // columns owned per workgroup (64*16 = 1024)
#define TPB     256                // 8 waves (wave32)
#define NWAVE   (TPB / 32)
#define KSPAN   (D_DIM / NWAVE)    // 128 reduction rows per wave
#define NCHUNK  (KSPAN / 4)        // 32 WMMA (K=4) per wave per step
#define NACC    4                  // independent accumulators (ILP)

#define PANEL_ELEMS (D_DIM * COLS)                 // 16384 floats = 64 KB
#define LDS_FLOATS  (2 * PANEL_ELEMS + D_DIM + NWAVE * COLS)
#define LDS_BYTES   (LDS_FLOATS * 4)               // 135,680 B < 320 KB/WGP

typedef float v2f  __attribute__((ext_vector_type(2)));
typedef float v8f  __attribute__((ext_vector_type(8)));
typedef unsigned int u32x4 __attribute__((ext_vector_type(4)));
typedef int   i32x4 __attribute__((ext_vector_type(4)));
typedef int   i32x8 __attribute__((ext_vector_type(8)));

#ifdef __has_builtin
#if __has_builtin(__builtin_amdgcn_tensor_load_to_lds)
#define HAVE_TDM 1
#endif
#if __has_builtin(__builtin_amdgcn_groupstaticsize)
#define HAVE_GSS 1
#endif
#endif

// ---------------------------------------------------------------------------
// Device-scope barrier: monotonic arrival counter in workspace, sense-free.
// Counter is zeroed by the init kernel every call, so replays are identical.
// ---------------------------------------------------------------------------
__device__ __forceinline__ void grid_barrier(unsigned* bar, unsigned step) {
    __syncthreads();
    if (threadIdx.x == 0) {
        __threadfence();  // release our v_new stores to device scope
        __hip_atomic_fetch_add(bar, 1u, __ATOMIC_ACQ_REL, __HIP_MEMORY_SCOPE_AGENT);
        const unsigned target = (step + 1u) * (unsigned)NWG;
        while (__hip_atomic_load(bar, __ATOMIC_ACQUIRE, __HIP_MEMORY_SCOPE_AGENT) < target) {
            __builtin_amdgcn_s_sleep(2);
        }
        __threadfence();  // acquire other WGs' v_new stores
    }
    __syncthreads();
}

// ---------------------------------------------------------------------------
// TDM: async-copy the 1024x16 fp32 column panel of core[xi] (cols j0..j0+15)
// into LDS at lds_byte_off. 2D descriptor per CDNA5 ISA:
//   tile_dim0 = 16 elems (64 B rows), tile_dim1 = 1024 rows,
//   tensor_dim0_stride = 1024 elems, data_size = 4 B, no pad/iterate/multicast.
// Tracked on TENSORcnt; consumed after s_wait_tensorcnt(0) + workgroup barrier.
// ---------------------------------------------------------------------------
__device__ __forceinline__ void tdm_prefetch_panel(const float* core, int xi,
                                                   int j0, unsigned lds_byte_off) {
#if defined(HAVE_TDM)
    const uint64_t gaddr = (uint64_t)(uintptr_t)core +
        ((uint64_t)xi * (uint64_t)(D_DIM * D_DIM) + (uint64_t)j0) * 4ull;

    u32x4 g0;
    g0[0] = 1u;                                            // count=1, user mode
    g0[1] = lds_byte_off;                                  // lds_addr (bytes)
    g0[2] = (unsigned)(gaddr & 0xFFFFFFFFu);               // global_addr[31:0]
    g0[3] = (unsigned)((gaddr >> 32) & 0x01FFFFFFu)        // global_addr[56:32]
          | (2u << 30);                                    // type = 2 (image)

    i32x8 g1;
    g1[0] = 0x00020000;                  // wg_mask=0 (not in cluster), data_size=4B
    g1[1] = (int)(COLS  << 16);          // tensor_dim0 = 16
    g1[2] = (int)(D_DIM << 16);          // tensor_dim1 = 1024
    g1[3] = (int)(COLS  << 16);          // tile_dim0 = 16 (64 B rows)
    g1[4] = D_DIM;                       // tile_dim1 = 1024, tile_dim2 = 0
    g1[5] = D_DIM;                       // tensor_dim0_stride = 1024 elems
    g1[6] = 0;                           // dim0_stride hi / dim1_stride lo16
    g1[7] = (D_DIM * D_DIM) >> 16;       // tensor_dim1_stride[47:16] (unused, 2D)

    const i32x4 z4 = {0, 0, 0, 0};
#if __clang_major__ >= 23
    const i32x8 z8 = {0, 0, 0, 0, 0, 0, 0, 0};
    __builtin_amdgcn_tensor_load_to_lds(g0, g1, z4, z4, z8, 0);
#else
    __builtin_amdgcn_tensor_load_to_lds(g0, g1, z4, z4, 0);
#endif
#else
    (void)core; (void)xi; (void)j0; (void)lds_byte_off;
#endif
}

__device__ __forceinline__ void wait_tensor0() {
#if defined(HAVE_TDM)
    __builtin_amdgcn_s_wait_tensorcnt(0);
#endif
}

#if !defined(HAVE_TDM)
// Fallback when the TDM builtin is unavailable: cooperative strided copy.
__device__ __forceinline__ void load_panel_direct(float* panel, const float* core,
                                                  int xi, int j0) {
    const float* src = core + (size_t)xi * D_DIM * D_DIM + j0;
    for (int r = threadIdx.x; r < D_DIM; r += TPB) {
        const float4 a = *(const float4*)(src + (size_t)r * D_DIM + 0);
        const float4 b = *(const float4*)(src + (size_t)r * D_DIM + 4);
        const float4 c = *(const float4*)(src + (size_t)r * D_DIM + 8);
        const float4 d = *(const float4*)(src + (size_t)r * D_DIM + 12);
        *(float4*)(panel + r * COLS + 0)  = a;
        *(float4*)(panel + r * COLS + 4)  = b;
        *(float4*)(panel + r * COLS + 8)  = c;
        *(float4*)(panel + r * COLS + 12) = d;
    }
}
#endif

// ---------------------------------------------------------------------------
// Init: zero barrier counter, seed v0 with the left boundary.
// ---------------------------------------------------------------------------
__global__ void tn_init_kernel(const float* __restrict__ left,
                               float* __restrict__ v0, unsigned* bar) {
    if (threadIdx.x == 0) *bar = 0u;
    for (int i = threadIdx.x; i < D_DIM; i += TPB) v0[i] = left[i];
}

// ---------------------------------------------------------------------------
// Persistent chain kernel. WG g owns columns [16g, 16g+16).
// Per step: WMMA-accumulate v_new[j] = sum_i v[i] * M[i,j] from the LDS panel,
// cross-wave reduce in LDS, publish to the global v double-buffer, barrier.
// ---------------------------------------------------------------------------
__global__ void __launch_bounds__(TPB)
tn_chain_kernel(const int* __restrict__ x, const float* __restrict__ core,
                float* __restrict__ vbuf0, float* __restrict__ vbuf1,
                unsigned* bar) {
    extern __shared__ float lds[];
    float* panelBase = lds;                         // 2 x 1024 x 16
    float* vsh       = lds + 2 * PANEL_ELEMS;       // 1024
    float* partial   = vsh + D_DIM;                 // NWAVE x 16

    const int j0   = (int)blockIdx.x * COLS;
    const int tid  = (int)threadIdx.x;
    const int wave = tid >> 5;
    const int lane = tid & 31;
    const bool hi  = (lane & 16) != 0;
    const int col  = lane & 15;

    unsigned dynBase = 0;
#if defined(HAVE_GSS)
    dynBase = __builtin_amdgcn_groupstaticsize();   // dynamic LDS base offset
#endif

    // Prologue: v_0 into LDS, panel for step 0 into buffer 0.
    for (int i = tid; i < D_DIM; i += TPB) vsh[i] = vbuf0[i];
#if defined(HAVE_TDM)
    if (wave == 0) tdm_prefetch_panel(core, x[0], j0, dynBase);
    wait_tensor0();
#else
    load_panel_direct(panelBase, core, x[0], j0);
#endif
    __syncthreads();

    for (int s = 0; s < F_LEN; ++s) {
        const int buf = s & 1;
        const float* pan = panelBase + buf * PANEL_ELEMS;
        float* vnext = (buf == 0) ? vbuf1 : vbuf0;

#if defined(HAVE_TDM)
        // Prefetch next step's panel into the other buffer while we compute.
        if (wave == 0 && (s + 1) < F_LEN) {
            tdm_prefetch_panel(core, x[s + 1], j0,
                               dynBase + (unsigned)((buf ^ 1) * PANEL_ELEMS * 4));
        }
#endif
        // GEMV tile via V_WMMA_F32_16X16X4_F32.
        //  A (16x4): every row = v[i0..i0+3]  -> VGPR0={K0|K2}, VGPR1={K1|K3}
        //  B (4x16): panel rows i0..i0+3      -> VGPR0={K0|K2}, VGPR1={K1|K3}
        //  C row 0, lanes 0..15 = partial dot for the 16 columns.
        // Four independent accumulators break the WMMA->WMMA RAW chain so the
        // scheduler can issue LDS loads for later chunks under earlier WMMAs.
        v8f acc[NACC];
        #pragma unroll
        for (int a = 0; a < NACC; ++a)
            acc[a] = (v8f){0.f, 0.f, 0.f, 0.f, 0.f, 0.f, 0.f, 0.f};

        const int ibase = wave * KSPAN;
        #pragma unroll 2
        for (int k = 0; k < NCHUNK; k += NACC) {
            #pragma unroll
            for (int a = 0; a < NACC; ++a) {
                const int i0 = ibase + 4 * (k + a);
                const int ia = i0 + (hi ? 2 : 0);   // ia is even -> 8B aligned
                const v2f av = *reinterpret_cast<const v2f*>(&vsh[ia]);
                v2f bv;
                bv[0] = pan[ia * COLS + col];
                bv[1] = pan[(ia + 1) * COLS + col];
                acc[a] = __builtin_amdgcn_wmma_f32_16x16x4_f32(
                             false, av, false, bv, (short)0, acc[a], false, false);
            }
        }
        const float r = (acc[0][0] + acc[1][0]) + (acc[2][0] + acc[3][0]);
        if (lane < 16) partial[wave * COLS + col] = r;
        __syncthreads();

        if (tid < COLS) {
            float sum = 0.f;
            #pragma unroll
            for (int w = 0; w < NWAVE; ++w) sum += partial[w * COLS + tid];
            vnext[j0 + tid] = sum;
        }

        grid_barrier(bar, (unsigned)s);             // all 64 WGs published v_{s+1}

        if (s + 1 < F_LEN) {
            for (int i = tid; i < D_DIM; i += TPB) vsh[i] = vnext[i];
#if defined(HAVE_TDM)
            wait_tensor0();                          // next panel landed in LDS
#else
            load_panel_direct(panelBase + (buf ^ 1) * PANEL_ELEMS,
                              core, x[s + 1], j0);
#endif
            __syncthreads();
        }
    }
}

// ---------------------------------------------------------------------------
// Final: out = dot(v_final, right). F is even so v_final lives in vbuf0.
// ---------------------------------------------------------------------------
__global__ void tn_dot_kernel(const float* __restrict__ v,
                              const float* __restrict__ right,
                              float* __restrict__ out) {
    __shared__ float red[TPB];
    float acc = 0.f;
    for (int i = threadIdx.x; i < D_DIM; i += TPB) acc += v[i] * right[i];
    red[threadIdx.x] = acc;
    __syncthreads();
    for (int s = TPB / 2; s > 0; s >>= 1) {
        if ((int)threadIdx.x < s) red[threadIdx.x] += red[threadIdx.x + s];
        __syncthreads();
    }
    if (threadIdx.x == 0) out[0] = red[0];
}

extern "C" void kernel_launch(void* const* d_in, const int* in_sizes, int n_in,
                              void* d_out, int out_size, void* d_ws, size_t ws_size,
                              hipStream_t stream) {
    const int*   x     = (const int*)d_in[0];     // (F,)
    const float* core  = (const float*)d_in[1];   // (d, D, D)
    const float* left  = (const float*)d_in[2];   // (D,)
    const float* right = (const float*)d_in[3];   // (D,)
    float* out = (float*)d_out;

    // Workspace: v double-buffer (2 x 4 KB) + barrier counter.
    float*    vbuf0 = (float*)d_ws;
    float*    vbuf1 = vbuf0 + D_DIM;
    unsigned* bar   = (unsigned*)(vbuf1 + D_DIM);

    tn_init_kernel<<<1, TPB, 0, stream>>>(left, vbuf0, bar);
    tn_chain_kernel<<<NWG, TPB, LDS_BYTES, stream>>>(x, core, vbuf0, vbuf1, bar);
    tn_dot_kernel<<<1, TPB, 0, stream>>>(vbuf0, right, out);
}